// Convolutional_Partial_Trace_28183575396357
// MI455X (gfx1250) — compile-verified
//
#include <hip/hip_runtime.h>

// D=2, N=13, Q=(3,7,11): partial trace of 8192x8192 f32 -> 8x8 f32.
// out[i,j] = sum_{t=0}^{1023} rho[spread(t)|place(i), spread(t)|place(j)]
// Kept bit positions (from LSB of 13-bit index): {10, 6, 2}.
// Traced positions: {12,11,9,8,7,5,4,3,1,0}; any bijection t->positions is valid.

typedef __attribute__((ext_vector_type(2))) float v2f;
typedef __attribute__((ext_vector_type(8))) float v8f;

__device__ __forceinline__ unsigned spread10(unsigned t) {
    // t bits 0..9 -> positions {0,1, 3,4,5, 7,8,9, 11,12}
    unsigned r = t & 0x3u;          // bits 0,1 -> pos 0,1
    r |= (t & 0x01Cu) << 1;         // bits 2,3,4 -> pos 3,4,5
    r |= (t & 0x0E0u) << 2;         // bits 5,6,7 -> pos 7,8,9
    r |= (t & 0x300u) << 3;         // bits 8,9  -> pos 11,12
    return r;
}

__device__ __forceinline__ unsigned place3(unsigned x) {
    // x bit0 -> pos 2 (qudit 11), bit1 -> pos 6 (qudit 7), bit2 -> pos 10 (qudit 3)
    return ((x & 1u) << 2) | ((x & 2u) << 5) | ((x & 4u) << 8);
}

__global__ void ptrace_zero_kernel(float* __restrict__ out) {
    if (threadIdx.x < 64) out[threadIdx.x] = 0.0f;
}

// Grid: 64 blocks x 32 threads (1 wave each).
// block = rtile*16 + s: rtile in [0,4) selects 16 rows of the 64-row G matrix,
// s in [0,16) selects a 64-wide slice of t. Each lane accumulates two f32
// partial sums (A-matrix 16x4 layout), one f32 WMMA with B=ones reduces K,
// lanes with N==0 atomically add 16 row sums into out.
__global__ void ptrace_wmma_kernel(const float* __restrict__ rho,
                                   float* __restrict__ out) {
    const int lane  = (int)threadIdx.x;   // 0..31
    const int w     = (int)blockIdx.x;    // 0..63
    const int rtile = w >> 4;             // 0..3
    const int s     = w & 15;             // t-slice 0..15

    const int m = lane & 15;              // A-matrix row within tile
    const int h = lane >> 4;              // lane half selects K pair {0,1} vs {2,3}

    const int g = rtile * 16 + m;         // flat output row: g = i*8 + j
    const unsigned i  = (unsigned)(g >> 3);
    const unsigned j  = (unsigned)(g & 7);
    const unsigned base = place3(i) * 8192u + place3(j);

    // Two f32 accumulators per lane = A[m, 2h] and A[m, 2h+1].
    v2f a;
    #pragma unroll
    for (int v = 0; v < 2; ++v) {
        const int kk = 2 * h + v;         // chunk 0..3 within this wave's slice
        float acc = 0.0f;
        #pragma unroll
        for (int u = 0; u < 16; ++u) {
            const unsigned t  = (unsigned)(s * 64 + kk * 16 + u);
            const unsigned sp = spread10(t);
            acc += rho[(size_t)sp * 8193u + base];
        }
        a[v] = acc;
    }

    // B = ones(4x16): every element is 1.0, so exact VGPR layout is irrelevant.
    v2f b; b[0] = 1.0f; b[1] = 1.0f;
    v8f c = {};

    // D[m,n] = sum_k A[m,k] for every column n (full f32 precision).
    // 8 args: (neg_a, A, neg_b, B, c_mod, C, reuse_a, reuse_b)
    v8f d = __builtin_amdgcn_wmma_f32_16x16x4_f32(
        /*neg_a=*/false, a, /*neg_b=*/false, b,
        /*c_mod=*/(short)0, c, /*reuse_a=*/false, /*reuse_b=*/false);

    // C/D layout: VGPR q, lanes 0-15 -> M=q, N=lane; lanes 16-31 -> M=8+q.
    // Column N=0 lives in lane 0 (M=0..7) and lane 16 (M=8..15).
    if ((lane & 15) == 0) {
        const int mbase = rtile * 16 + (h ? 8 : 0);
        #pragma unroll
        for (int q = 0; q < 8; ++q) {
            atomicAdd(&out[mbase + q], d[q]);
        }
    }
}

extern "C" void kernel_launch(void* const* d_in, const int* in_sizes, int n_in,
                              void* d_out, int out_size, void* d_ws, size_t ws_size,
                              hipStream_t stream) {
    const float* rho = (const float*)d_in[0];   // 8192*8192 f32
    float* out = (float*)d_out;                 // 64 f32

    (void)in_sizes; (void)n_in; (void)out_size; (void)d_ws; (void)ws_size;

    hipLaunchKernelGGL(ptrace_zero_kernel, dim3(1), dim3(64), 0, stream, out);
    hipLaunchKernelGGL(ptrace_wmma_kernel, dim3(64), dim3(32), 0, stream, rho, out);
}